// GCNLayer_81217831568021
// MI455X (gfx1250) — compile-verified
//
#include <hip/hip_runtime.h>
#include <hip/hip_bf16.h>

typedef __attribute__((ext_vector_type(2))) float v2f;
typedef __attribute__((ext_vector_type(8))) float v8f;

#define D 128  // D_IN == D_OUT == 128

// ---------------------------------------------------------------------------
// Kernel 1: out[n, d] = bias[d]   (fold bias into the scatter target init)
// ---------------------------------------------------------------------------
__global__ __launch_bounds__(256) void gcn_bias_init(const float* __restrict__ bias,
                                                     float* __restrict__ out,
                                                     int total) {
    int idx = blockIdx.x * 256 + threadIdx.x;
    if (idx < total) out[idx] = bias[idx & (D - 1)];
}

// ---------------------------------------------------------------------------
// Kernel 2: support = X @ W  via V_WMMA_F32_16X16X4_F32 (full fp32 precision).
// Block = 256 threads = 8 waves. Block b owns M-rows [16b, 16b+16); wave w
// owns N-cols [16w, 16w+16). K loop of 128 stepped by 4 -> 32 WMMAs/wave.
//
// Fragment layouts (ISA 7.12.2, 32-bit, wave32):
//   A 16x4 : lane L<16 -> M=L,    v0=K0, v1=K1 ; lane L>=16 -> M=L-16, v0=K2, v1=K3
//   B 4x16 : lane L<16 -> N=L,    v0=K0, v1=K1 ; lane L>=16 -> N=L-16, v0=K2, v1=K3
//   C 16x16: VGPR r, lanes 0-15 -> (M=r, N=lane); lanes 16-31 -> (M=r+8, N=lane-16)
// ---------------------------------------------------------------------------
__global__ __launch_bounds__(256) void gcn_gemm(const float* __restrict__ X,
                                                const float* __restrict__ W,
                                                float* __restrict__ S,
                                                int n_nodes) {
    const int lane   = threadIdx.x & 31;
    const int wave   = threadIdx.x >> 5;      // 0..7 -> N tile
    const int m_base = blockIdx.x * 16;
    if (m_base >= n_nodes) return;            // uniform guard, EXEC stays all-1s
    const int n_base = wave * 16;

    const int half = lane >> 4;               // 0: K pair {0,1}, 1: K pair {2,3}
    const int lid  = lane & 15;

    const float* __restrict__ arow = X + (size_t)(m_base + lid) * D;  // A row
    const float* __restrict__ bcol = W + (n_base + lid);              // B column

    v8f acc = {};
#pragma unroll
    for (int k = 0; k < D; k += 4) {
        const int ka = k + half * 2;
        v2f a, b;
        a.x = arow[ka + 0];                    // consecutive -> global_load_b64
        a.y = arow[ka + 1];
        b.x = bcol[(size_t)(ka + 0) * D];      // stride-D rows of W
        b.y = bcol[(size_t)(ka + 1) * D];
        acc = __builtin_amdgcn_wmma_f32_16x16x4_f32(
            /*neg_a=*/false, a, /*neg_b=*/false, b,
            /*c_mod=*/(short)0, acc, /*reuse_a=*/false, /*reuse_b=*/false);
    }

    float* __restrict__ srow = S + (size_t)(m_base + half * 8) * D + (n_base + lid);
#pragma unroll
    for (int r = 0; r < 8; ++r)
        srow[(size_t)r * D] = acc[r];
}

// ---------------------------------------------------------------------------
// Kernel 3: out[dst] += vals[e] * support[src]. One wave per edge (grid-stride);
// lane handles 4 consecutive features via a 16B gather + 4 f32 atomics.
// support (51.2 MB) fits in the 192 MB L2, so random gathers hit L2 and the
// atomics resolve in L2's atomic units.
// ---------------------------------------------------------------------------
__global__ __launch_bounds__(256) void gcn_scatter(const float* __restrict__ S,
                                                   const int* __restrict__ esrc,
                                                   const int* __restrict__ edst,
                                                   const float* __restrict__ evals,
                                                   float* __restrict__ out,
                                                   int n_edges) {
    const int lane    = threadIdx.x & 31;
    const int wave_g  = blockIdx.x * (256 >> 5) + (threadIdx.x >> 5);
    const int n_waves = gridDim.x * (256 >> 5);

    for (int e = wave_g; e < n_edges; e += n_waves) {
        const int   s = esrc[e];
        const int   d = edst[e];
        const float v = evals[e];

        const float4 g = *((const float4*)(S + (size_t)s * D) + lane);
        float* op = out + (size_t)d * D + lane * 4;
        atomicAdd(op + 0, g.x * v);
        atomicAdd(op + 1, g.y * v);
        atomicAdd(op + 2, g.z * v);
        atomicAdd(op + 3, g.w * v);
    }
}

// ---------------------------------------------------------------------------
extern "C" void kernel_launch(void* const* d_in, const int* in_sizes, int n_in,
                              void* d_out, int out_size, void* d_ws, size_t ws_size,
                              hipStream_t stream) {
    const float* X     = (const float*)d_in[0];   // [N, 128]
    const int*   esrc  = (const int*)  d_in[1];   // [E]
    const int*   edst  = (const int*)  d_in[2];   // [E]
    const float* evals = (const float*)d_in[3];   // [E]
    const float* W     = (const float*)d_in[4];   // [128, 128]
    const float* bias  = (const float*)d_in[5];   // [128]
    float*       out   = (float*)d_out;           // [N, 128]
    float*       S     = (float*)d_ws;            // support scratch [N, 128]

    const int n_nodes = in_sizes[0] / D;
    const int n_edges = in_sizes[1];
    const int total   = n_nodes * D;

    gcn_bias_init<<<(total + 255) / 256, 256, 0, stream>>>(bias, out, total);
    gcn_gemm<<<(n_nodes + 15) / 16, 256, 0, stream>>>(X, W, S, n_nodes);
    gcn_scatter<<<8192, 256, 0, stream>>>(S, esrc, edst, evals, out, n_edges);
}